// TopKSparse_70360154243700
// MI455X (gfx1250) — compile-verified
//
#include <hip/hip_runtime.h>
#include <hip/hip_bf16.h>
#include <stdint.h>

#define DIM        2048
#define KSEL       512          // log2(16)/16 * 2048
#define THREADS    256
#define CHUNKS     (DIM / 4)    // float4 chunks per row
#define RAND_P     0.1f

typedef unsigned int u32x4_t __attribute__((ext_vector_type(4)));
typedef int          i32x8_t __attribute__((ext_vector_type(8)));
typedef int          i32x4_t __attribute__((ext_vector_type(4)));
typedef float        f32x4_t __attribute__((ext_vector_type(4)));

#if defined(__HIP_DEVICE_COMPILE__)
  #if __has_builtin(__builtin_amdgcn_tensor_load_to_lds)
    #define USE_TDM 1
  #else
    #define USE_TDM 0
  #endif
  #if __has_builtin(__builtin_nontemporal_store)
    #define NT_STORE(v, p) __builtin_nontemporal_store((v), (p))
  #else
    #define NT_STORE(v, p) (*(p) = (v))
  #endif
#else
  #define USE_TDM 0
  #define NT_STORE(v, p) (*(p) = (v))
#endif

#if USE_TDM
// Issue one TDM descriptor moving a 2048-element f32 row (8 KB) into LDS.
// D# layout per CDNA5 ISA ch.7.3/8: group0 = {flags, lds_addr, global_addr, type},
// group1 = {data_size, tensor dims/strides, tile dims}. Groups 2/3 unused (2D).
__device__ __forceinline__ void tdm_load_row_f32(unsigned lds_addr, const float* gsrc) {
    const unsigned long long ga = (unsigned long long)(size_t)gsrc;
    u32x4_t g0;
    g0.x = 1u;                                            // count=1, user descriptor
    g0.y = lds_addr;                                      // LDS byte address
    g0.z = (unsigned)ga;                                  // global_addr[31:0]
    g0.w = (unsigned)((ga >> 32) & 0x01FFFFFFull)         // global_addr[56:32]
         | 0x80000000u;                                   // type=2 ("image") in [127:126]
    i32x8_t g1;
    g1[0] = (int)(2u << 16);                              // data_size=2 (4 bytes), wg_mask=0
    g1[1] = (int)((unsigned)(DIM & 0xFFFF) << 16);        // tensor_dim0[15:0] @ bits[63:48]
    g1[2] = (int)(1u << 16);                              // tensor_dim1[15:0]=1 @ bits[95:80]
    g1[3] = (int)((unsigned)(DIM & 0xFFFF) << 16);        // tile_dim0=2048 @ bits[127:112]
    g1[4] = 1;                                            // tile_dim1=1, tile_dim2=0
    g1[5] = DIM;                                          // tensor_dim0_stride low32
    g1[6] = (int)((unsigned)(DIM & 0xFFFF) << 16);        // tensor_dim1_stride[15:0] @ [223:208]
    g1[7] = 0;
    i32x4_t gz = {0, 0, 0, 0};
#if __clang_major__ >= 23
    i32x8_t gz8 = {0, 0, 0, 0, 0, 0, 0, 0};
    __builtin_amdgcn_tensor_load_to_lds(g0, g1, gz, gz, gz8, 0);
#else
    __builtin_amdgcn_tensor_load_to_lds(g0, g1, gz, gz, 0);
#endif
}
#endif // USE_TDM

__global__ __launch_bounds__(THREADS)
void TopKSparse_70360154243700_kernel(const float* __restrict__ x,
                                      const float* __restrict__ rnd,
                                      float* __restrict__ out) {
    __shared__ __align__(16) float s_x[DIM];   // 8 KB: row of x
    __shared__ __align__(16) float s_r[DIM];   // 8 KB: row of rand_u
    __shared__ unsigned s_hist[256];           // radix histogram
    __shared__ unsigned s_gsum[32];            // 8-bin group sums
    __shared__ unsigned s_meta[2];             // [0]=prefix bits, [1]=remaining k
    __shared__ unsigned s_count;               // selected count

    const int tid = threadIdx.x;
    const size_t row = blockIdx.x;
    const float* xrow = x   + row * (size_t)DIM;
    const float* rrow = rnd + row * (size_t)DIM;
    float*       orow = out + row * (size_t)DIM;

#if USE_TDM
    if (tid == 0) {
        tdm_load_row_f32((unsigned)(size_t)(void*)s_x, xrow);
        tdm_load_row_f32((unsigned)(size_t)(void*)s_r, rrow);
        // Wait INSIDE the issuing branch so it cannot be scheduled before the
        // loads (round-1 codegen hoisted a builtin wait into the else-leg).
        asm volatile("s_wait_tensorcnt 0x0" ::: "memory");
    }
#else
    {
        const f32x4_t* xs = (const f32x4_t*)xrow;
        const f32x4_t* rs = (const f32x4_t*)rrow;
        f32x4_t* sx4 = (f32x4_t*)s_x;
        f32x4_t* sr4 = (f32x4_t*)s_r;
        for (int c = tid; c < CHUNKS; c += THREADS) { sx4[c] = xs[c]; sr4[c] = rs[c]; }
    }
#endif
    if (tid == 0) { s_meta[0] = 0u; s_meta[1] = KSEL; s_count = 0u; }
    __syncthreads();

    const int c0 = tid, c1 = tid + THREADS;    // two float4 chunks per thread

    // ---- 4-round radix select: bit pattern of the K-th largest |x| ----
    for (int r = 0; r < 4; ++r) {
        const int shift = 24 - 8 * r;
        s_hist[tid] = 0u;
        __syncthreads();
        const unsigned prefix  = s_meta[0];
        const unsigned hi_mask = (r == 0) ? 0u : (0xFFFFFFFFu << (shift + 8));
        #pragma unroll
        for (int p = 0; p < 2; ++p) {
            const int c = p ? c1 : c0;
            const f32x4_t v = ((const f32x4_t*)s_x)[c];
            #pragma unroll
            for (int j = 0; j < 4; ++j) {
                const unsigned key = __float_as_uint(v[j]) & 0x7FFFFFFFu;
                if ((key & hi_mask) == prefix)
                    atomicAdd(&s_hist[(key >> shift) & 0xFFu], 1u);
            }
        }
        __syncthreads();
        // Two-level descending scan: 32 parallel group sums, then <=40 serial steps.
        if (tid < 32) {
            unsigned s = 0;
            #pragma unroll
            for (int j = 0; j < 8; ++j) s += s_hist[tid * 8 + j];
            s_gsum[tid] = s;
        }
        __syncthreads();
        if (tid == 0) {
            const unsigned kk = s_meta[1];
            unsigned cum = 0u;
            int g = 31;
            for (; g >= 1; --g) {
                const unsigned c = s_gsum[g];
                if (cum + c >= kk) break;
                cum += c;
            }
            int b = g * 8 + 7;
            for (; b > g * 8; --b) {
                const unsigned c = s_hist[b];
                if (cum + c >= kk) break;
                cum += c;
            }
            s_meta[0] = prefix | ((unsigned)b << shift);
            s_meta[1] = kk - cum;
        }
        __syncthreads();
    }
    const unsigned thresh = s_meta[0];

    // ---- mask = topk | (rand < p); count selected ----
    unsigned mbits = 0u, cnt = 0u;
    #pragma unroll
    for (int p = 0; p < 2; ++p) {
        const int c = p ? c1 : c0;
        const f32x4_t v = ((const f32x4_t*)s_x)[c];
        const f32x4_t u = ((const f32x4_t*)s_r)[c];
        #pragma unroll
        for (int j = 0; j < 4; ++j) {
            const unsigned key = __float_as_uint(v[j]) & 0x7FFFFFFFu;
            const bool m = (key >= thresh) || (u[j] < RAND_P);
            mbits |= (unsigned)m << (p * 4 + j);
            cnt += (unsigned)m;
        }
    }
    atomicAdd(&s_count, cnt);
    __syncthreads();
    const float scale = (float)DIM / fmaxf((float)s_count, 1.0f);

    // ---- scaled, masked write-back (coalesced non-temporal b128 stores) ----
    #pragma unroll
    for (int p = 0; p < 2; ++p) {
        const int c = p ? c1 : c0;
        const f32x4_t v = ((const f32x4_t*)s_x)[c];
        f32x4_t o;
        #pragma unroll
        for (int j = 0; j < 4; ++j)
            o[j] = ((mbits >> (p * 4 + j)) & 1u) ? v[j] * scale : 0.0f;
        NT_STORE(o, (f32x4_t*)orow + c);
    }
}

extern "C" void kernel_launch(void* const* d_in, const int* in_sizes, int n_in,
                              void* d_out, int out_size, void* d_ws, size_t ws_size,
                              hipStream_t stream) {
    const float* x = (const float*)d_in[0];
    const float* r = (const float*)d_in[1];
    float* out = (float*)d_out;
    const int M = in_sizes[0] / DIM;   // 8*4096 = 32768 rows
    TopKSparse_70360154243700_kernel<<<dim3(M), dim3(THREADS), 0, stream>>>(x, r, out);
}